// VideoSetCluster2FormerCriterion_87497073754795
// MI455X (gfx1250) — compile-verified
//
#include <hip/hip_runtime.h>

// ---------------------------------------------------------------------------
// VideoSetCluster2FormerCriterion for MI455X (gfx1250, wave32)
//
// loss = CE(pred_logits vs scattered targets, weighted) +
//        weighted cosine-pair loss over 500K gathered row pairs.
//
// Pair loss is L2-gather bound (160MB table resident in 192MB L2; 800MB of
// gathered reads). We keep f32 precision and use V_WMMA_F32_16X16X4_F32 to
// compute diag(A*Bt), diag(A*At), diag(B*Bt) for 16 pairs per wave: for the
// f32 WMMA shape the A-operand register image of a tile equals the B-operand
// image of its transpose, so the same registers feed SRC0/SRC1.
//
// K-slot mapping exploits dot-product permutation invariance: lanes 0-15 own
// row elements [0,100), lanes 16-31 own [100,200), so each b128 load supplies
// two consecutive WMMA K-steps (halves load-instruction count vs b64).
// ---------------------------------------------------------------------------

constexpr int kNumClasses = 100;
constexpr int kC1 = kNumClasses + 1;   // 101
constexpr int kQ = 200;                // cluster feature dim
constexpr float kEps = 1e-8f;
constexpr float kMinPairW = 0.1f;

typedef __attribute__((ext_vector_type(8))) float v8f;
typedef __attribute__((ext_vector_type(4))) float v4f;
typedef __attribute__((ext_vector_type(2))) float v2f;

__device__ __forceinline__ float vsel8(v8f v, int i) {
  float r = v[0];
  if (i == 1) r = v[1];
  if (i == 2) r = v[2];
  if (i == 3) r = v[3];
  if (i == 4) r = v[4];
  if (i == 5) r = v[5];
  if (i == 6) r = v[6];
  if (i == 7) r = v[7];
  return r;
}

__device__ __forceinline__ float waveSum32(float v) {
#pragma unroll
  for (int o = 16; o > 0; o >>= 1) v += __shfl_xor(v, o, 32);
  return v;
}

__device__ __forceinline__ float waveMax32(float v) {
#pragma unroll
  for (int o = 16; o > 0; o >>= 1) v = fmaxf(v, __shfl_xor(v, o, 32));
  return v;
}

// --- ws init: acc[0..3] = 0, target_classes[] = NUM_CLASSES -----------------
__global__ void init_ws_kernel(float* __restrict__ acc, int* __restrict__ tclass,
                               int nrows) {
  const int i = blockIdx.x * blockDim.x + threadIdx.x;
  if (i < 4) acc[i] = 0.0f;
  if (i < nrows) tclass[i] = kNumClasses;
}

// --- scatter matched targets ------------------------------------------------
__global__ void scatter_kernel(const int* __restrict__ targets,
                               const int* __restrict__ idx_b,
                               const int* __restrict__ idx_q,
                               int* __restrict__ tclass, int M) {
  const int m = blockIdx.x * blockDim.x + threadIdx.x;
  if (m < M) tclass[idx_b[m] * kQ + idx_q[m]] = targets[m];
}

// --- CE loss: one wave32 per (b,q) row of 101 logits ------------------------
__global__ void ce_loss_kernel(const float* __restrict__ logits,
                               const int* __restrict__ tclass,
                               const float* __restrict__ ew,
                               float* __restrict__ acc, int nrows) {
  const int lane = threadIdx.x & 31;
  const int row = (blockIdx.x * blockDim.x + threadIdx.x) >> 5;
  if (row >= nrows) return;
  const float* rowp = logits + (size_t)row * kC1;

  float m = -3.402823466e38f;
  for (int j = lane; j < kC1; j += 32) m = fmaxf(m, rowp[j]);
  m = waveMax32(m);

  float s = 0.0f;
  for (int j = lane; j < kC1; j += 32) s += expf(rowp[j] - m);
  s = waveSum32(s);

  if (lane == 0) {
    const int t = tclass[row];
    const float w = ew[t];
    const float nll = (m + logf(s)) - rowp[t];  // -(logit[t] - logsumexp)
    atomicAdd(&acc[0], w * nll);
    atomicAdd(&acc[1], w);
  }
}

// --- pair loss: 16 pairs per wave via V_WMMA_F32_16X16X4_F32 ----------------
__global__ void cluster_loss_kernel(const float* __restrict__ src,
                                    const int* __restrict__ pairs,
                                    const unsigned char* __restrict__ pair_pos,
                                    const float* __restrict__ pair_w,
                                    float* __restrict__ acc, int P) {
  __shared__ float s_c[8];
  __shared__ float s_w[8];
  const int lane = threadIdx.x & 31;
  const int waveInBlock = threadIdx.x >> 5;
  const int wave = blockIdx.x * (blockDim.x >> 5) + waveInBlock;
  const int base = wave * 16;  // 16 pairs per wave tile

  float contrib = 0.0f, wsum = 0.0f;
  if (base < P) {  // wave-uniform guard: EXEC all-1s around WMMA
    const int p = lane & 15;
    const int g = min(base + p, P - 1);
    const int ia = pairs[2 * g];
    const int ib = pairs[2 * g + 1];
    // Lanes 0-15 own elements [0,100) of their pair's rows (K slots 0,1);
    // lanes 16-31 own elements [100,200) (K slots 2,3). Valid because the
    // reduction over K is order-invariant and A/B use identical mappings.
    const int region = (lane >> 4) * (kQ / 2);
    const float* ap = src + (size_t)ia * kQ + region;
    const float* bp = src + (size_t)ib * kQ + region;

    v8f cab = {};
    v8f caa = {};
    v8f cbb = {};
    for (int i = 0; i < kQ / 2; i += 4) {
      const v4f a4 = *(const v4f*)(ap + i);   // b128: feeds two K-steps
      const v4f b4 = *(const v4f*)(bp + i);
      const v2f a01 = {a4.x, a4.y};
      const v2f b01 = {b4.x, b4.y};
      cab = __builtin_amdgcn_wmma_f32_16x16x4_f32(false, a01, false, b01,
                                                  (short)0, cab, false, false);
      caa = __builtin_amdgcn_wmma_f32_16x16x4_f32(false, a01, false, a01,
                                                  (short)0, caa, false, false);
      cbb = __builtin_amdgcn_wmma_f32_16x16x4_f32(false, b01, false, b01,
                                                  (short)0, cbb, false, false);
      const v2f a23 = {a4.z, a4.w};
      const v2f b23 = {b4.z, b4.w};
      cab = __builtin_amdgcn_wmma_f32_16x16x4_f32(false, a23, false, b23,
                                                  (short)0, cab, false, false);
      caa = __builtin_amdgcn_wmma_f32_16x16x4_f32(false, a23, false, a23,
                                                  (short)0, caa, false, false);
      cbb = __builtin_amdgcn_wmma_f32_16x16x4_f32(false, b23, false, b23,
                                                  (short)0, cbb, false, false);
    }

    // Diagonal (p,p): p<8 -> lane p, VGPR p; p>=8 -> lane p+16, VGPR p-8.
    const bool active = (lane < 8) || (lane >= 24);
    if (active) {
      const int pid = (lane < 8) ? lane : (lane - 16);
      const int vi = (lane < 8) ? lane : (lane - 24);
      const int gg = base + pid;
      if (gg < P) {
        const float dab = vsel8(cab, vi);
        const float daa = vsel8(caa, vi);
        const float dbb = vsel8(cbb, vi);
        const float w = fmaxf(pair_w[gg], kMinPairW);
        const float c = dab / (sqrtf(daa) * sqrtf(dbb) + kEps);
        const float per = pair_pos[gg] ? (1.0f - c) : fmaxf(c, 0.0f);
        contrib = w * per;
        wsum = w;
      }
    }
  }

  contrib = waveSum32(contrib);
  wsum = waveSum32(wsum);
  if (lane == 0) {
    s_c[waveInBlock] = contrib;
    s_w[waveInBlock] = wsum;
  }
  __syncthreads();
  if (threadIdx.x == 0) {
    float tc = 0.0f, tw = 0.0f;
    const int nw = blockDim.x >> 5;
    for (int i = 0; i < nw; ++i) {
      tc += s_c[i];
      tw += s_w[i];
    }
    atomicAdd(&acc[2], tc);
    atomicAdd(&acc[3], tw);
  }
}

// --- finalize ---------------------------------------------------------------
__global__ void finalize_kernel(const float* __restrict__ acc,
                                float* __restrict__ out) {
  if (threadIdx.x == 0 && blockIdx.x == 0)
    out[0] = acc[0] / acc[1] + acc[2] / acc[3];
}

extern "C" void kernel_launch(void* const* d_in, const int* in_sizes, int n_in,
                              void* d_out, int out_size, void* d_ws,
                              size_t ws_size, hipStream_t stream) {
  const float* pred_logits = (const float*)d_in[0];          // (B,Q,101) f32
  const int* targets = (const int*)d_in[1];                  // (M,) i32
  const int* idx_b = (const int*)d_in[2];                    // (M,) i32
  const int* idx_q = (const int*)d_in[3];                    // (M,) i32
  const float* src_clusters = (const float*)d_in[4];         // (NPTS,Q) f32
  const int* pairs = (const int*)d_in[5];                    // (P,2) i32
  const unsigned char* pair_pos = (const unsigned char*)d_in[6];  // (P,) bool
  const float* pair_weights = (const float*)d_in[7];         // (P,) f32
  const float* empty_weight = (const float*)d_in[8];         // (101,) f32

  const int M = in_sizes[1];
  const int P = in_sizes[7];
  const int nrows = in_sizes[0] / kC1;  // B*Q

  float* acc = (float*)d_ws;                    // 4 accumulators
  int* tclass = (int*)((char*)d_ws + 16);       // B*Q target classes

  init_ws_kernel<<<(nrows + 255) / 256, 256, 0, stream>>>(acc, tclass, nrows);
  scatter_kernel<<<(M + 255) / 256, 256, 0, stream>>>(targets, idx_b, idx_q,
                                                      tclass, M);
  ce_loss_kernel<<<(nrows * 32 + 255) / 256, 256, 0, stream>>>(
      pred_logits, tclass, empty_weight, acc, nrows);

  const int tiles = (P + 15) / 16;                 // 16 pairs per wave
  const int blocks = (tiles + 7) / 8;              // 8 waves per 256-thr block
  cluster_loss_kernel<<<blocks, 256, 0, stream>>>(src_clusters, pairs, pair_pos,
                                                  pair_weights, acc, P);
  finalize_kernel<<<1, 1, 0, stream>>>(acc, (float*)d_out);
}